// MatryoshkaSoftTopKLoss_71021579207126
// MI455X (gfx1250) — compile-verified
//
#include <hip/hip_runtime.h>
#include <hip/hip_bf16.h>
#include <math.h>

#define B_DIM  512
#define T_DIM  256
#define D_DIM  128
#define TILE_T 64
#define TSTR   132          // padded LDS row stride in floats (528B, 16B aligned)
#define NSCALE 3
#define CLAMP_V 50.0f

typedef __attribute__((ext_vector_type(2))) float v2f;
typedef __attribute__((ext_vector_type(8))) float v8f;
typedef __attribute__((ext_vector_type(4))) int   v4i;
typedef __attribute__((ext_vector_type(4))) float v4f;

typedef __attribute__((address_space(1))) v4i* gv4p;   // global int4*
typedef __attribute__((address_space(3))) v4i* lv4p;   // LDS int4*

__device__ __forceinline__ void async_copy_b128(const void* gsrc, void* ldst) {
    // AS1 pointer value == flat address; AS3 pointer value == low 32 bits of flat LDS address
    __builtin_amdgcn_global_load_async_to_lds_b128(
        (gv4p)(uintptr_t)gsrc,
        (lv4p)(unsigned int)(uintptr_t)ldst,
        0, 0);
}

__global__ __launch_bounds__(256) void matryoshka_soft_topk_kernel(
    const float* __restrict__ emb,          // (B,T,D)
    const float* __restrict__ w,            // (D)
    const float* __restrict__ bias,         // (1)
    const float* __restrict__ temp,         // (1)
    const unsigned char* __restrict__ mask, // (B,T) bool
    float* __restrict__ out)                // (3,B,T,D)
{
    __shared__ float sE[TILE_T * TSTR];     // 33792 B tile
    __shared__ float sW[D_DIM];
    __shared__ float sScore[T_DIM];
    __shared__ float sGate[NSCALE][T_DIM];
    __shared__ float sThr[NSCALE];

    const int tid  = threadIdx.x;
    const int lane = tid & 31;
    const int wave = tid >> 5;
    const int b    = blockIdx.x;

    if (tid < D_DIM) sW[tid] = w[tid];

    const size_t rowBase = (size_t)b * T_DIM * D_DIM;

    // ---------- Phase 1: scores via async global->LDS tiles + WMMA f32 16x16x4 ----------
    for (int t0 = 0; t0 < T_DIM; t0 += TILE_T) {
        __syncthreads();   // all readers of previous tile are done before overwrite
        // copy TILE_T x 128 floats = 2048 float4s; 8 per thread, coalesced b128 async
        #pragma unroll
        for (int i = 0; i < 8; ++i) {
            int idx = i * 256 + tid;              // 0..2047
            int tt  = idx >> 5;                   // token within tile
            int f4  = idx & 31;                   // float4 within row
            const float* gs = emb + rowBase + (size_t)(t0 + tt) * D_DIM + f4 * 4;
            float*       ls = &sE[tt * TSTR + f4 * 4];
            async_copy_b128(gs, ls);
        }
        __builtin_amdgcn_s_wait_asynccnt(0);
        __syncthreads();

        if (wave < 4) {
            // wave g handles tokens [t0 + 16g, t0 + 16g + 16)
            const int g   = wave;
            const int row = g * 16 + (lane & 15);     // row in tile
            const int kb  = (lane >> 4) << 1;         // K sub-offset: 0 (lanes 0-15) / 2 (lanes 16-31)
            v8f c = {};
            #pragma unroll
            for (int k0 = 0; k0 < D_DIM; k0 += 4) {
                v2f a, bb;
                // A (16x4 f32): V0 = K=0|K=2, V1 = K=1|K=3 per lane half
                a.x = sE[row * TSTR + k0 + kb];
                a.y = sE[row * TSTR + k0 + kb + 1];
                // B (4x16 f32): every column = w chunk -> every C column = dot(E_row, w)
                bb.x = sW[k0 + kb];
                bb.y = sW[k0 + kb + 1];
                c = __builtin_amdgcn_wmma_f32_16x16x4_f32(
                        false, a, false, bb, (short)0, c, false, false);
            }
            // C/D f32 16x16 layout: VGPR v = row M=v (lanes 0-15) / M=v+8 (lanes 16-31)
            if (lane == 0) {
                #pragma unroll
                for (int v = 0; v < 8; ++v) sScore[t0 + g * 16 + v] = c[v];
            } else if (lane == 16) {
                #pragma unroll
                for (int v = 0; v < 8; ++v) sScore[t0 + g * 16 + 8 + v] = c[v];
            }
        }
    }
    __syncthreads();

    // ---------- Phase 2: bias + mask(-inf), rank-select the 3 thresholds ----------
    const float bv = bias[0];
    const float tv = temp[0];
    const float mf = mask[(size_t)b * T_DIM + tid] ? 1.0f : 0.0f;
    {
        float s = sScore[tid] + bv;
        if (mf == 0.0f) s = -INFINITY;
        sScore[tid] = s;                       // each thread owns its slot
    }
    __syncthreads();
    {
        const float st = sScore[tid];
        int cnt = 0;
        for (int j = 0; j < T_DIM; ++j) {      // broadcast reads, tiny
            float sj = sScore[j];
            cnt += (sj > st) || (sj == st && j < tid);
        }
        if (cnt == 127) sThr[0] = st;          // k = 128  (k-th largest = rank k-1)
        if (cnt == 63)  sThr[1] = st;          // k = 64
        if (cnt == 31)  sThr[2] = st;          // k = 32
    }
    __syncthreads();

    // ---------- Phase 2b: gates (nan->-50, clip +/-50, sigmoid * mask) ----------
    {
        const float st = sScore[tid];
        #pragma unroll
        for (int s = 0; s < NSCALE; ++s) {
            float d = st - sThr[s];
            if (d != d) d = -CLAMP_V;                       // nan_to_num
            d = fminf(fmaxf(d, -CLAMP_V), CLAMP_V);         // clip
            sGate[s][tid] = mf / (1.0f + __expf(-d * tv));  // sigmoid * mask
        }
    }
    __syncthreads();

    // ---------- Phase 3: stream embeddings (L2-resident re-read), 3 gated NT stores ----------
    // Output planes (201 MB) are write-once: use non-temporal stores so they don't
    // evict the L2-resident embeddings (67 MB) being re-read here.
    const size_t planeStride = (size_t)B_DIM * T_DIM * D_DIM;
    const v4f* __restrict__ ev = (const v4f*)(emb + rowBase);
    for (int i = tid; i < T_DIM * (D_DIM / 4); i += 256) {
        const int t = i >> 5;                  // 32 float4 per token row
        v4f e = ev[i];
        #pragma unroll
        for (int s = 0; s < NSCALE; ++s) {
            const float g = sGate[s][t];
            v4f o = e * g;
            __builtin_nontemporal_store(o, (v4f*)(out + (size_t)s * planeStride + rowBase) + i);
        }
    }
}

extern "C" void kernel_launch(void* const* d_in, const int* in_sizes, int n_in,
                              void* d_out, int out_size, void* d_ws, size_t ws_size,
                              hipStream_t stream) {
    (void)in_sizes; (void)n_in; (void)out_size; (void)d_ws; (void)ws_size;
    const float*         emb  = (const float*)d_in[0];
    const float*         w    = (const float*)d_in[1];
    const float*         bias = (const float*)d_in[2];
    const float*         temp = (const float*)d_in[3];
    const unsigned char* mask = (const unsigned char*)d_in[4];
    float*               out  = (float*)d_out;

    dim3 grid(B_DIM), block(256);
    hipLaunchKernelGGL(matryoshka_soft_topk_kernel, grid, block, 0, stream,
                       emb, w, bias, temp, mask, out);
}